// IoULoss_17712445129189
// MI455X (gfx1250) — compile-verified
//
#include <hip/hip_runtime.h>

#ifndef __has_builtin
#define __has_builtin(x) 0
#endif

#define EPS_F 1e-7f
#define THREADS 256
#define BOXES_PER_BLOCK 1024
#define TILE_FLOATS (BOXES_PER_BLOCK * 3)   // 3072 floats = 12 KB per array
#define CHUNKS (TILE_FLOATS / 4)            // 768 x b128 chunks per array

typedef int v4i __attribute__((ext_vector_type(4)));
typedef __attribute__((address_space(1))) v4i* global_v4i_ptr;
typedef __attribute__((address_space(3))) v4i* lds_v4i_ptr;

// ---- CDNA5 async global->LDS staging (ASYNCcnt path), with safe fallback ----
__device__ __forceinline__ void stage16(const float* __restrict__ gsrc,
                                        float* __restrict__ ldst) {
#if __has_builtin(__builtin_amdgcn_global_load_async_to_lds_b128)
  __builtin_amdgcn_global_load_async_to_lds_b128(
      (global_v4i_ptr)(gsrc),
      (lds_v4i_ptr)(ldst),
      /*imm offset=*/0, /*cpol=*/0);
#else
  // Fallback: still b128 global load + ds_store_b128 (coalesced staging)
  *(float4*)ldst = *(const float4*)gsrc;
#endif
}

__device__ __forceinline__ void wait_async0() {
#if __has_builtin(__builtin_amdgcn_s_wait_asynccnt)
  __builtin_amdgcn_s_wait_asynccnt(0);
#else
  asm volatile("s_wait_asynccnt 0" ::: "memory");
#endif
}

__global__ void __launch_bounds__(THREADS)
iou_loss_main(const float* __restrict__ outs, const float* __restrict__ tars,
              float* __restrict__ partials, int nboxes) {
  // [0, 3072): outputs tile; [3072, 6144): targets tile. 24 KB total.
  __shared__ float tile[2 * TILE_FLOATS];

  const int t = threadIdx.x;
  const long long b0 = (long long)blockIdx.x * BOXES_PER_BLOCK; // first box of tile
  const long long gbase = b0 * 3;                                // float index
  const long long lim = 3LL * nboxes - 4;                        // last valid b128 base

  // Stage: 768 coalesced 16B chunks per array; 3 per thread per array.
#pragma unroll
  for (int k = 0; k < 3; ++k) {
    const int c = t + k * THREADS;                 // chunk 0..767
    long long gi = gbase + (long long)c * 4;       // float index of chunk
    if (gi > lim) gi = lim;                        // clamp (tail safety; unused data)
    stage16(outs + gi, &tile[c * 4]);
    stage16(tars + gi, &tile[TILE_FLOATS + c * 4]);
  }
  wait_async0();       // this wave's async loads complete
  __syncthreads();     // all waves waited -> entire tile visible in LDS

  float sum = 0.0f;
#pragma unroll
  for (int k = 0; k < 4; ++k) {
    const int i = t + k * THREADS;                 // box within tile
    if (b0 + i < (long long)nboxes) {
      const float ox = tile[3 * i + 0];
      const float oy = tile[3 * i + 1];
      const float os = tile[3 * i + 2];
      const float tx = tile[TILE_FLOATS + 3 * i + 0];
      const float ty = tile[TILE_FLOATS + 3 * i + 1];
      const float ts = tile[TILE_FLOATS + 3 * i + 2];

      const float ro = ox + os, lo = ox - os;
      const float bo = oy + os, to_ = oy - os;
      const float rt = tx + ts, lt = tx - ts;
      const float bt = ty + ts, tt = ty - ts;

      // xs[1]-xs[2] of sorted 4 values == -| min(pair maxes) - max(pair mins) |
      const float wx = fminf(fmaxf(ro, lo), fmaxf(rt, lt)) -
                       fmaxf(fminf(ro, lo), fminf(rt, lt));
      const float wy = fminf(fmaxf(bo, to_), fmaxf(bt, tt)) -
                       fmaxf(fminf(bo, to_), fminf(bt, tt));
      float ov = fabsf(wx) * fabsf(wy);

      const bool sep = (ro < lt) || (bo < tt) || (rt < lo) || (bt < to_);
      ov = sep ? 0.0f : ov;

      const float d_o = 2.0f * os, d_t = 2.0f * ts;   // match (2*sz)^2 rounding
      const float uni = d_o * d_o + d_t * d_t - ov;
      sum += 1.0f - ov / (uni + EPS_F);
    }
  }

  // Deterministic block reduction (reuse tile LDS after barrier).
  __syncthreads();
  tile[t] = sum;
  __syncthreads();
#pragma unroll
  for (int s = THREADS / 2; s > 0; s >>= 1) {
    if (t < s) tile[t] += tile[t + s];
    __syncthreads();
  }
  if (t == 0) partials[blockIdx.x] = tile[0];
}

__global__ void __launch_bounds__(THREADS)
iou_loss_final(const float* __restrict__ partials, float* __restrict__ out,
               int nblocks) {
  __shared__ float s[THREADS];
  float sum = 0.0f;
  for (int i = threadIdx.x; i < nblocks; i += THREADS) sum += partials[i];
  s[threadIdx.x] = sum;
  __syncthreads();
#pragma unroll
  for (int st = THREADS / 2; st > 0; st >>= 1) {
    if (threadIdx.x < st) s[threadIdx.x] += s[threadIdx.x + st];
    __syncthreads();
  }
  if (threadIdx.x == 0) out[0] = s[0];
}

extern "C" void kernel_launch(void* const* d_in, const int* in_sizes, int n_in,
                              void* d_out, int out_size, void* d_ws, size_t ws_size,
                              hipStream_t stream) {
  const float* outs = (const float*)d_in[0];
  const float* tars = (const float*)d_in[1];
  const int nboxes = in_sizes[0] / 3;                         // 4,194,304
  const int nblocks = (nboxes + BOXES_PER_BLOCK - 1) / BOXES_PER_BLOCK; // 4096

  float* partials = (float*)d_ws;                             // 16 KB scratch

  iou_loss_main<<<nblocks, THREADS, 0, stream>>>(outs, tars, partials, nboxes);
  iou_loss_final<<<1, THREADS, 0, stream>>>(partials, (float*)d_out, nblocks);
}